// TAU_25503515803936
// MI455X (gfx1250) — compile-verified
//
#include <hip/hip_runtime.h>
#include <hip/hip_bf16.h>

typedef __attribute__((ext_vector_type(2))) float v2f;
typedef __attribute__((ext_vector_type(8))) float v8f;

#define M_CENTERS 256
// NORM_CONST = 1 / ((2*pi)^(4/2) * 0.1^4) = 1 / ((2pi)^2 * 1e-4)
#define NORM_CONST_D 253.30295910584444
#define INV_2SIG2    50.0   // 0.5 / 0.1^2

// ---------------------------------------------------------------------------
// RBF net: net_out[n] = NORM_CONST * sum_m exp(-50*||x_n - c_m||^2) * wc_m
// One wave (32 lanes) handles a 16-row tile of x against all 256 centers,
// using V_WMMA_F32_16X16X4_F32 (K=4 == D) for the x.c cross terms with an
// f64->f32 hi/lo split (3 WMMAs per 16x16 tile) for compensated accuracy.
// ---------------------------------------------------------------------------
__global__ __launch_bounds__(256) void TAU_rbf_wmma_kernel(
    const double* __restrict__ x,        // N x 4
    const double* __restrict__ centers,  // 256 x 4
    const double* __restrict__ coeff,    // 256
    double* __restrict__ net_out,        // N
    int n)
{
    const int lane = threadIdx.x & 31;
    const int wave = blockIdx.x * (blockDim.x >> 5) + (threadIdx.x >> 5);
    const int row_base = wave * 16;
    if (row_base >= n) return;           // wave-uniform exit: EXEC stays all-1s

    const int half = lane >> 4;          // 0: K=0,1 / rows 0-7 ; 1: K=2,3 / rows 8-15
    const int l16  = lane & 15;

    // ---- coefficient mean (wave-redundant, 256 values) ----
    double csum = 0.0;
    #pragma unroll
    for (int i = 0; i < M_CENTERS / 32; ++i) csum += coeff[lane + 32 * i];
    #pragma unroll
    for (int m = 16; m >= 1; m >>= 1) csum += __shfl_xor(csum, m, 32);
    const double cmean = csum * (1.0 / M_CENTERS);

    // ---- A tile: 16 rows of x, split into f32 hi/lo per ISA A-layout ----
    int row = row_base + l16;
    if (row >= n) row = n - 1;           // clamp (keeps EXEC uniform; guarded store)
    const int k0 = half * 2;
    const double xa0 = x[row * 4 + k0 + 0];
    const double xa1 = x[row * 4 + k0 + 1];
    const float ah0 = (float)xa0, ah1 = (float)xa1;
    const float al0 = (float)(xa0 - (double)ah0);
    const float al1 = (float)(xa1 - (double)ah1);
    v2f Ah = {ah0, ah1};
    v2f Al = {al0, al1};

    // row norms ||x||^2 in f64; both wave halves get the full sum
    const double xn2p = xa0 * xa0 + xa1 * xa1;
    const double xn2  = xn2p + __shfl_xor(xn2p, 16, 32);
    // each lane accumulates rows (half*8 + j); gather those 8 row norms
    double xr[8];
    #pragma unroll
    for (int j = 0; j < 8; ++j)
        xr[j] = __shfl(xn2, half * 8 + j, 32);

    double s[8];
    #pragma unroll
    for (int j = 0; j < 8; ++j) s[j] = 0.0;

    // ---- sweep 16 column-tiles of the 256 centers ----
    #pragma unroll 1
    for (int ct = 0; ct < M_CENTERS / 16; ++ct) {
        const int col = ct * 16 + l16;
        const double cb0 = centers[col * 4 + k0 + 0];
        const double cb1 = centers[col * 4 + k0 + 1];
        const float bh0 = (float)cb0, bh1 = (float)cb1;
        const float bl0 = (float)(cb0 - (double)bh0);
        const float bl1 = (float)(cb1 - (double)bh1);
        v2f Bh = {bh0, bh1};
        v2f Bl = {bl0, bl1};

        const double cn2p = cb0 * cb0 + cb1 * cb1;
        const double cn2  = cn2p + __shfl_xor(cn2p, 16, 32);
        const double wcn  = coeff[col] - cmean;

        // cross[m][n] = x_m . c_n via 3 compensated K=4 WMMAs
        v8f zero = {};
        v8f acc  = __builtin_amdgcn_wmma_f32_16x16x4_f32(
                       false, Ah, false, Bh, (short)0, zero, false, false);
        v8f corr = __builtin_amdgcn_wmma_f32_16x16x4_f32(
                       false, Al, false, Bh, (short)0, zero, false, false);
        corr     = __builtin_amdgcn_wmma_f32_16x16x4_f32(
                       false, Ah, false, Bl, (short)0, corr, false, false);

        // D layout: VGPR j holds row (half*8 + j), column = l16
        #pragma unroll
        for (int j = 0; j < 8; ++j) {
            const double cross = (double)acc[j] + (double)corr[j];
            const double dist  = xr[j] + cn2 - 2.0 * cross;
            s[j] += wcn * exp(-INV_2SIG2 * dist);
        }
    }

    // ---- reduce over the 16 columns (xor masks stay within each half) ----
    #pragma unroll
    for (int m = 1; m <= 8; m <<= 1) {
        #pragma unroll
        for (int j = 0; j < 8; ++j)
            s[j] += __shfl_xor(s[j], m, 32);
    }

    if (l16 == 0) {
        #pragma unroll
        for (int j = 0; j < 8; ++j) {
            const int ridx = row_base + half * 8 + j;
            if (ridx < n) net_out[ridx] = NORM_CONST_D * s[j];
        }
    }
}

// ---------------------------------------------------------------------------
// ensemble[n] = probs[n,:] . w  +  norm_probs[n] * (1 - sum(w)) ; E = 10
// Purely memory bound (~8.8 MB) — plain coalesced kernel.
// ---------------------------------------------------------------------------
__global__ __launch_bounds__(256) void TAU_ensemble_kernel(
    const double* __restrict__ probs,    // N x 10
    const double* __restrict__ nprobs,   // N x 1
    const double* __restrict__ w,        // 10
    double* __restrict__ out,            // N
    int n)
{
    const int i = blockIdx.x * blockDim.x + threadIdx.x;
    if (i >= n) return;
    double wsum = 0.0, acc = 0.0;
    #pragma unroll
    for (int e = 0; e < 10; ++e) {
        const double we = w[e];          // uniform -> scalar loads
        wsum += we;
        acc  += probs[i * 10 + e] * we;
    }
    out[i] = acc + nprobs[i] * (1.0 - wsum);
}

extern "C" void kernel_launch(void* const* d_in, const int* in_sizes, int n_in,
                              void* d_out, int out_size, void* d_ws, size_t ws_size,
                              hipStream_t stream) {
    (void)n_in; (void)d_ws; (void)ws_size; (void)out_size;
    const double* x       = (const double*)d_in[0];  // N x 4
    const double* probs   = (const double*)d_in[1];  // N x 10
    const double* nprobs  = (const double*)d_in[2];  // N x 1
    const double* w       = (const double*)d_in[3];  // 10
    const double* centers = (const double*)d_in[4];  // 256 x 4
    const double* coeff   = (const double*)d_in[5];  // 256

    const int n = in_sizes[0] / 4;                   // N
    double* out_ensemble = (double*)d_out;           // first N elements
    double* out_net      = (double*)d_out + n;       // next N elements

    TAU_ensemble_kernel<<<(n + 255) / 256, 256, 0, stream>>>(
        probs, nprobs, w, out_ensemble, n);

    const int tiles  = (n + 15) / 16;                // one wave per 16-row tile
    const int wpb    = 256 / 32;                     // 8 waves per block
    const int blocks = (tiles + wpb - 1) / wpb;
    TAU_rbf_wmma_kernel<<<blocks, 256, 0, stream>>>(
        x, centers, coeff, out_net, n);
}